// SATLayer_18829136625864
// MI455X (gfx1250) — compile-verified
//
#include <hip/hip_runtime.h>
#include <hip/hip_bf16.h>
#include <math.h>

typedef __attribute__((ext_vector_type(2))) float v2f;
typedef __attribute__((ext_vector_type(8))) float v8f;

#define DIN  256
#define DOUT 64
#define NEG_SLOPE 0.01f

// ---------------------------------------------------------------------------
// Init: zero output + denom, set rowmax = -inf. Must run every call since the
// aggregation accumulates with atomics.
// ---------------------------------------------------------------------------
__global__ void gat_init(float* __restrict__ out, float* __restrict__ rowmax,
                         float* __restrict__ denom, int nOut, int nNodes) {
  int i = blockIdx.x * blockDim.x + threadIdx.x;
  if (i < nOut) out[i] = 0.0f;
  if (i < nNodes) { rowmax[i] = -__builtin_inff(); denom[i] = 0.0f; }
}

// ---------------------------------------------------------------------------
// h = X @ W + b  using fp32 WMMA (V_WMMA_F32_16X16X4_F32).
// One wave computes a 16(M) x 64(N) tile: 4 v8f accumulators, K looped by 4.
// A 16x4 fp32 layout: lanes 0-15 -> M=0..15; VGPR0 holds K=0 (lo half) / K=2
// (hi half), VGPR1 holds K=1 / K=3. B 4x16 mirrors that K split.
// ---------------------------------------------------------------------------
__global__ void gat_gemm_wmma(const float* __restrict__ X, const float* __restrict__ W,
                              const float* __restrict__ bias, float* __restrict__ H,
                              int nNodes) {
  const int wave = (int)((blockIdx.x * blockDim.x + threadIdx.x) >> 5);
  const int lane = (int)(threadIdx.x & 31);
  const int m0 = wave * 16;
  if (m0 >= nNodes) return;                 // wave-uniform: EXEC stays all-ones
  const int halfSel = lane >> 4;            // 0: K+0/K+1, 1: K+2/K+3
  const int ncol = lane & 15;
  int mrow = m0 + (lane & 15);
  if (mrow >= nNodes) mrow = nNodes - 1;    // safe clamp (N is a multiple of 16)

  const float* __restrict__ xrow = X + (size_t)mrow * DIN;
  v8f acc0 = {}, acc1 = {}, acc2 = {}, acc3 = {};

  for (int k = 0; k < DIN; k += 4) {
    const int kb = k + 2 * halfSel;
    v2f a;
    a.x = xrow[kb];
    a.y = xrow[kb + 1];
    const float* __restrict__ w0 = W + (size_t)kb * DOUT;  // row kb of W
    const float* __restrict__ w1 = w0 + DOUT;              // row kb+1
    v2f b0, b1, b2, b3;
    b0.x = w0[ncol];      b0.y = w1[ncol];
    b1.x = w0[16 + ncol]; b1.y = w1[16 + ncol];
    b2.x = w0[32 + ncol]; b2.y = w1[32 + ncol];
    b3.x = w0[48 + ncol]; b3.y = w1[48 + ncol];
    acc0 = __builtin_amdgcn_wmma_f32_16x16x4_f32(false, a, false, b0, (short)0, acc0, false, false);
    acc1 = __builtin_amdgcn_wmma_f32_16x16x4_f32(false, a, false, b1, (short)0, acc1, false, false);
    acc2 = __builtin_amdgcn_wmma_f32_16x16x4_f32(false, a, false, b2, (short)0, acc2, false, false);
    acc3 = __builtin_amdgcn_wmma_f32_16x16x4_f32(false, a, false, b3, (short)0, acc3, false, false);
  }

  const float bv0 = bias[ncol];
  const float bv1 = bias[16 + ncol];
  const float bv2 = bias[32 + ncol];
  const float bv3 = bias[48 + ncol];
#pragma unroll
  for (int j = 0; j < 8; ++j) {
    const int r = m0 + j + 8 * halfSel;     // C/D layout: VGPR j -> M=j / M=j+8
    if (r < nNodes) {
      float* __restrict__ hr = H + (size_t)r * DOUT;
      hr[ncol]      = acc0[j] + bv0;
      hr[16 + ncol] = acc1[j] + bv1;
      hr[32 + ncol] = acc2[j] + bv2;
      hr[48 + ncol] = acc3[j] + bv3;
    }
  }
}

// ---------------------------------------------------------------------------
// a1[i] = h[i] . a1_w + a1_b ; a2[i] = h[i] . a2_w + a2_b. One wave per node.
// ---------------------------------------------------------------------------
__global__ void gat_node_att(const float* __restrict__ H,
                             const float* __restrict__ a1w, const float* __restrict__ a1b,
                             const float* __restrict__ a2w, const float* __restrict__ a2b,
                             float* __restrict__ a1, float* __restrict__ a2, int nNodes) {
  const int wave = (int)((blockIdx.x * blockDim.x + threadIdx.x) >> 5);
  const int lane = (int)(threadIdx.x & 31);
  if (wave >= nNodes) return;
  const float2 hv = ((const float2*)(H + (size_t)wave * DOUT))[lane];
  const float2 w1 = ((const float2*)a1w)[lane];
  const float2 w2 = ((const float2*)a2w)[lane];
  float s1 = hv.x * w1.x + hv.y * w1.y;
  float s2 = hv.x * w2.x + hv.y * w2.y;
#pragma unroll
  for (int off = 16; off > 0; off >>= 1) {
    s1 += __shfl_down(s1, off, 32);
    s2 += __shfl_down(s2, off, 32);
  }
  if (lane == 0) {
    a1[wave] = s1 + a1b[0];
    a2[wave] = s2 + a2b[0];
  }
}

// ---------------------------------------------------------------------------
// Per-edge: score = LeakyReLU(a1[row] + a2[col]); segmented max via sign-split
// integer-punned float atomic max/min (rowmax pre-initialized to -inf).
// ---------------------------------------------------------------------------
__global__ void gat_edge_score(const int* __restrict__ rows, const int* __restrict__ cols,
                               const float* __restrict__ a1, const float* __restrict__ a2,
                               float* __restrict__ scores, float* __restrict__ rowmax, int E) {
  const int e = blockIdx.x * blockDim.x + threadIdx.x;
  if (e >= E) return;
  const int r = rows[e];
  const int c = cols[e];
  float s = a1[r] + a2[c];
  s = (s > 0.0f) ? s : NEG_SLOPE * s;
  scores[e] = s;
  if (s >= 0.0f)
    atomicMax((int*)(rowmax + r), __float_as_int(s));
  else
    atomicMin((unsigned int*)(rowmax + r), __float_as_uint(s));
}

// ---------------------------------------------------------------------------
// Per-edge: ex = exp(score - rowmax[row]); scores <- ex; denom[row] += ex.
// ---------------------------------------------------------------------------
__global__ void gat_edge_exp(const int* __restrict__ rows, float* __restrict__ scores,
                             const float* __restrict__ rowmax, float* __restrict__ denom, int E) {
  const int e = blockIdx.x * blockDim.x + threadIdx.x;
  if (e >= E) return;
  const int r = rows[e];
  const float ex = expf(scores[e] - rowmax[r]);
  scores[e] = ex;
  atomicAdd(denom + r, ex);
}

// ---------------------------------------------------------------------------
// out[row] += (ex / denom[row]) * h[col]. One wave per edge, float2 per lane.
// h (12.8 MB) and out (12.8 MB) both stay resident in the 192 MB L2, so the
// 205 MB of gathered traffic and 51 M f32 atomics resolve at L2 rates.
// ---------------------------------------------------------------------------
__global__ void gat_aggregate(const int* __restrict__ rows, const int* __restrict__ cols,
                              const float* __restrict__ H, const float* __restrict__ ex,
                              const float* __restrict__ denom, float* __restrict__ out, int E) {
  const int wave = (int)((blockIdx.x * blockDim.x + threadIdx.x) >> 5);
  const int lane = (int)(threadIdx.x & 31);
  if (wave >= E) return;
  const int r = rows[wave];
  const int c = cols[wave];
  const float att = ex[wave] / denom[r];
  const float2 hv = ((const float2*)(H + (size_t)c * DOUT))[lane];
  float* o = out + (size_t)r * DOUT + lane * 2;
  atomicAdd(o,     att * hv.x);
  atomicAdd(o + 1, att * hv.y);
}

// ---------------------------------------------------------------------------
extern "C" void kernel_launch(void* const* d_in, const int* in_sizes, int n_in,
                              void* d_out, int out_size, void* d_ws, size_t ws_size,
                              hipStream_t stream) {
  const float* features = (const float*)d_in[0];
  const int*   ei       = (const int*)d_in[1];
  // d_in[2] = edge_values (computed-but-unused in the reference)
  const float* W   = (const float*)d_in[3];
  const float* b   = (const float*)d_in[4];
  const float* a1w = (const float*)d_in[5];
  const float* a1b = (const float*)d_in[6];
  const float* a2w = (const float*)d_in[7];
  const float* a2b = (const float*)d_in[8];
  float* out = (float*)d_out;

  const int N = in_sizes[0] / DIN;      // 50000
  const int E = in_sizes[1] / 2;        // 800000
  const int* rows = ei;
  const int* cols = ei + E;

  // Workspace layout (floats): H[N*64] | a1[N] | a2[N] | rowmax[N] | denom[N] | scores[E]
  float* ws     = (float*)d_ws;
  float* H      = ws;
  float* a1     = H + (size_t)N * DOUT;
  float* a2     = a1 + N;
  float* rowmax = a2 + N;
  float* denom  = rowmax + N;
  float* scores = denom + N;

  const int nOut = N * DOUT;
  const int T = 256;

  gat_init<<<(nOut + T - 1) / T, T, 0, stream>>>(out, rowmax, denom, nOut, N);

  {
    const int nWaves = (N + 15) / 16;                  // 3125 tiles
    const int grid = (nWaves * 32 + T - 1) / T;
    gat_gemm_wmma<<<grid, T, 0, stream>>>(features, W, b, H, N);
  }
  {
    const long long thr = (long long)N * 32;           // wave per node
    gat_node_att<<<(int)((thr + T - 1) / T), T, 0, stream>>>(H, a1w, a1b, a2w, a2b, a1, a2, N);
  }
  gat_edge_score<<<(E + T - 1) / T, T, 0, stream>>>(rows, cols, a1, a2, scores, rowmax, E);
  gat_edge_exp  <<<(E + T - 1) / T, T, 0, stream>>>(rows, scores, rowmax, denom, E);
  {
    const long long thr = (long long)E * 32;           // wave per edge
    gat_aggregate<<<(int)((thr + T - 1) / T), T, 0, stream>>>(rows, cols, H, scores, denom, out, E);
  }
}